// MVTEModel_19061064859838
// MI455X (gfx1250) — compile-verified
//
#include <hip/hip_runtime.h>
#include <hip/hip_bf16.h>

typedef __attribute__((ext_vector_type(2))) float v2f;
typedef __attribute__((ext_vector_type(8))) float v8f;

#define DIM 256
#define LDA 260   // padded LDS row stride (260 % 64 == 4 -> conflict-free row reads)

// ---------------------------------------------------------------------------
// Scatter-add: one block per edge, 256 threads = feature dim.
// accum[dst, d] += src[s, d];  cnt[dst] += 1 (thread 0)
// ---------------------------------------------------------------------------
__global__ void mvte_scatter_add(const float* __restrict__ src,
                                 const int*   __restrict__ sidx,
                                 const int*   __restrict__ didx,
                                 float*       __restrict__ accum,
                                 float*       __restrict__ cnt,
                                 int nedges)
{
    int e = blockIdx.x;
    if (e >= nedges) return;
    int d = threadIdx.x;
    int s   = sidx[e];
    int dst = didx[e];
    float v = src[(size_t)s * DIM + d];
    atomicAdd(&accum[(size_t)dst * DIM + d], v);
    if (d == 0) atomicAdd(&cnt[dst], 1.0f);
}

// ---------------------------------------------------------------------------
// O = act( (A ./ max(cnt,1)) @ W^T + bias ),  A,O: [M,256], W: [256,256] row-major.
// Block = 256 threads (8 waves), one 16-row M tile per block, each wave does
// two 16-col N tiles with V_WMMA_F32_16X16X4_F32.  O may alias A (rows are
// staged into LDS before the in-place store).
// ---------------------------------------------------------------------------
__global__ void mvte_gemm(const float* __restrict__ A,
                          const float* __restrict__ cnt,   // nullptr => no mean division
                          const float* __restrict__ W,
                          const float* __restrict__ bias,
                          float*       __restrict__ O,
                          int relu)
{
    __shared__ float Alds[16 * LDA];
    const int tid  = threadIdx.x;
    const int wave = tid >> 5;
    const int lane = tid & 31;
    const int hv   = lane >> 4;    // half-wave: selects K pair (A) / row group (D)
    const int l    = lane & 15;
    const int m0   = blockIdx.x * 16;

    // Cooperatively stage scaled A tile (16 x 256) into LDS.
    for (int i = tid; i < 16 * DIM; i += 256) {
        int r = i >> 8;
        int c = i & 255;
        float s = 1.0f;
        if (cnt) s = 1.0f / fmaxf(cnt[m0 + r], 1.0f);
        Alds[r * LDA + c] = A[(size_t)(m0 + r) * DIM + c] * s;
    }
    __syncthreads();

    const int n0 = wave * 32;                       // this wave: cols [n0, n0+32)
    v8f c0 = {0.f, 0.f, 0.f, 0.f, 0.f, 0.f, 0.f, 0.f};
    v8f c1 = c0;
    const float* Wr0 = W + (size_t)(n0 + l)      * DIM;  // B[k][n] = W[n][k]
    const float* Wr1 = W + (size_t)(n0 + 16 + l) * DIM;

    #pragma unroll 4
    for (int k0 = 0; k0 < DIM; k0 += 4) {
        const int k = k0 + 2 * hv;
        v2f a, b0, b1;
        // A 16x4 layout: lanes 0-15 hold K=k0,k0+1; lanes 16-31 hold K=k0+2,k0+3
        a.x  = Alds[l * LDA + k];
        a.y  = Alds[l * LDA + k + 1];
        b0.x = Wr0[k];  b0.y = Wr0[k + 1];
        b1.x = Wr1[k];  b1.y = Wr1[k + 1];
        c0 = __builtin_amdgcn_wmma_f32_16x16x4_f32(false, a, false, b0, (short)0, c0, false, false);
        c1 = __builtin_amdgcn_wmma_f32_16x16x4_f32(false, a, false, b1, (short)0, c1, false, false);
    }

    // D layout: VGPR v -> row m0 + v + 8*hv, col n0 + l
    const float bn0 = bias[n0 + l];
    const float bn1 = bias[n0 + 16 + l];
    #pragma unroll
    for (int v = 0; v < 8; ++v) {
        int row  = m0 + v + 8 * hv;
        float o0 = c0[v] + bn0;
        float o1 = c1[v] + bn1;
        if (relu) { o0 = fmaxf(o0, 0.f); o1 = fmaxf(o1, 0.f); }
        O[(size_t)row * DIM + n0 + l]      = o0;
        O[(size_t)row * DIM + n0 + 16 + l] = o1;
    }
}

// ---------------------------------------------------------------------------
// Fused: w = softmax(alpha); V[x] = w0*E[x] + w1*vtsum[x]/max(cnt[x],1);
// out[b] = gamma - || V[h] + R[r] - V[t] ||.  One block (256 thr) per triple.
// ---------------------------------------------------------------------------
__global__ void mvte_score(const float* __restrict__ E,
                           const float* __restrict__ R,
                           const float* __restrict__ vtsum,
                           const float* __restrict__ cnt_ent,
                           const int*   __restrict__ triples,
                           const float* __restrict__ alpha,
                           const float* __restrict__ gamma,
                           float*       __restrict__ out)
{
    __shared__ float part[8];
    const int b = blockIdx.x;
    const int d = threadIdx.x;
    const int h = triples[b * 3 + 0];
    const int r = triples[b * 3 + 1];
    const int t = triples[b * 3 + 2];

    float a0 = alpha[0], a1 = alpha[1];
    float mx = fmaxf(a0, a1);
    float e0 = expf(a0 - mx), e1 = expf(a1 - mx);
    float inv = 1.0f / (e0 + e1);
    float w0 = e0 * inv, w1 = e1 * inv;

    float ch = fmaxf(cnt_ent[h], 1.0f);
    float ct = fmaxf(cnt_ent[t], 1.0f);
    float vh = w0 * E[(size_t)h * DIM + d] + w1 * vtsum[(size_t)h * DIM + d] / ch;
    float vt = w0 * E[(size_t)t * DIM + d] + w1 * vtsum[(size_t)t * DIM + d] / ct;
    float diff = vh + R[(size_t)r * DIM + d] - vt;
    float s = diff * diff;

    // wave32 reduction, then 8 wave partials via LDS
    for (int o = 16; o > 0; o >>= 1) s += __shfl_xor(s, o, 32);
    if ((d & 31) == 0) part[d >> 5] = s;
    __syncthreads();
    if (d == 0) {
        float tot = 0.f;
        for (int i = 0; i < 8; ++i) tot += part[i];
        out[b] = gamma[0] - sqrtf(tot);
    }
}

// ---------------------------------------------------------------------------
extern "C" void kernel_launch(void* const* d_in, const int* in_sizes, int n_in,
                              void* d_out, int out_size, void* d_ws, size_t ws_size,
                              hipStream_t stream)
{
    (void)in_sizes; (void)n_in; (void)out_size; (void)ws_size;

    const float* E      = (const float*)d_in[0];
    const float* R      = (const float*)d_in[1];
    const float* lin2_w = (const float*)d_in[2];
    const float* lin2_b = (const float*)d_in[3];
    const float* lin3_w = (const float*)d_in[4];
    const float* lin3_b = (const float*)d_in[5];
    const float* t2e_w  = (const float*)d_in[6];
    const float* t2e_b  = (const float*)d_in[7];
    const float* alpha  = (const float*)d_in[8];
    const float* gamma  = (const float*)d_in[9];
    const int*   et     = (const int*)d_in[10];   // [2, 900000]
    const int*   tt     = (const int*)d_in[11];   // [2, 600000]
    const int*   etet   = (const int*)d_in[12];   // [2, 600000]
    const int*   trip   = (const int*)d_in[13];   // [8192, 3]

    const int NE  = 100000, NTRI = 300000, NTET = 150000;
    const int ETE = 900000, TTE  = 600000, ETTE = 600000, B = 8192;

    // Workspace layout (floats). v_topo region aliases the (dead-by-then)
    // triangle buffer: stream ordering makes this safe.
    float* ws      = (float*)d_ws;
    float* buf_tri = ws;                             // 76.8M  (tri sums -> tri_emb)
    float* buf_vt  = ws;                             // 25.6M  (aliases buf_tri)
    float* cnt_ent = ws + (size_t)NE * DIM;          // 0.1M   (inside buf_tri region)
    float* cnt_tri = ws + (size_t)NTRI * DIM;        // 0.3M
    float* buf_tet = cnt_tri + NTRI;                 // 38.4M  (tet sums -> tet_emb -> tet_proj)
    float* cnt_tet = buf_tet + (size_t)NTET * DIM;   // 0.15M

    // 1) triangles <- mean over entity-triangle edges
    hipMemsetAsync(buf_tri, 0, (size_t)NTRI * DIM * sizeof(float), stream);
    hipMemsetAsync(cnt_tri, 0, (size_t)NTRI * sizeof(float), stream);
    mvte_scatter_add<<<ETE, 256, 0, stream>>>(E, et, et + ETE, buf_tri, cnt_tri, ETE);

    // 2) tri_emb = relu(mean @ lin2^T + b)  (in place, mean folded in)
    mvte_gemm<<<NTRI / 16, 256, 0, stream>>>(buf_tri, cnt_tri, lin2_w, lin2_b, buf_tri, 1);

    // 3) tetras <- mean over triangle-tetra edges
    hipMemsetAsync(buf_tet, 0, (size_t)NTET * DIM * sizeof(float), stream);
    hipMemsetAsync(cnt_tet, 0, (size_t)NTET * sizeof(float), stream);
    mvte_scatter_add<<<TTE, 256, 0, stream>>>(buf_tri, tt, tt + TTE, buf_tet, cnt_tet, TTE);

    // 4) tet_emb = relu(mean @ lin3^T + b);  5) tet_proj = tet_emb @ t2e^T + b
    mvte_gemm<<<NTET / 16, 256, 0, stream>>>(buf_tet, cnt_tet, lin3_w, lin3_b, buf_tet, 1);
    mvte_gemm<<<NTET / 16, 256, 0, stream>>>(buf_tet, nullptr, t2e_w,  t2e_b,  buf_tet, 0);

    // 6) v_topo sums <- tet_proj over entity-tetra edges (src = row1, dst = row0)
    hipMemsetAsync(buf_vt,  0, (size_t)NE * DIM * sizeof(float), stream);
    hipMemsetAsync(cnt_ent, 0, (size_t)NE * sizeof(float), stream);
    mvte_scatter_add<<<ETTE, 256, 0, stream>>>(buf_tet, etet + ETTE, etet, buf_vt, cnt_ent, ETTE);

    // 7) fused blend + TransE scoring
    mvte_score<<<B, 256, 0, stream>>>(E, R, buf_vt, cnt_ent, trip, alpha, gamma, (float*)d_out);
}